// QuantumLayer_34076270526775
// MI455X (gfx1250) — compile-verified
//
#include <hip/hip_runtime.h>
#include <math.h>

// out[b][q] = prod_{j<=q} cos(pi * clip(in[b][j],0,1))   for b in [0,16384), q in [0,12)
//
// Implemented as a log-domain prefix-sum via v_wmma_f32_16x16x4_f32:
//   D_log = Log|cos|(16x16, cols 12..15 = 0) x UpperTriOnes(16x16)   -> sum of logs
//   D_cnt = [cos<0](16x16)                   x UpperTriOnes(16x16)   -> #negatives (exact)
// out = (-1)^{cnt} * exp(sum_log).  One wave32 owns one 16-row tile.

typedef float v2f __attribute__((ext_vector_type(2)));
typedef float v8f __attribute__((ext_vector_type(8)));

#define NQ    12
#define BATCH 16384

__global__ __launch_bounds__(256) void qlayer_wmma_kernel(const float* __restrict__ in,
                                                          float* __restrict__ out) {
    const int lane = threadIdx.x & 31;
    const int wave = threadIdx.x >> 5;
    const int tile = blockIdx.x * 8 + wave;   // 8 waves/block, 16 rows/tile
    const int row0 = tile * 16;
    const int m    = lane & 15;               // A-matrix row (and B/C/D column N) for this lane
    const int hi   = (lane >= 16) ? 1 : 0;
    const int row  = row0 + m;

    // ---- load & transform the 6 qubit values this lane owns (A 16x4 layout) ----
    // low half lanes own j = 4*kc + {0,1}; high half lanes own j = 4*kc + {2,3}
    // each pair is contiguous -> one aligned float2 load per K-chunk
    const float2* rp2 = (const float2*)(in + row * NQ);

    float lg[6];  // log|cos(pi x)| (clamped: no -inf, so tri-zeros never make NaN)
    float ng[6];  // 1.0f if cos(pi x) < 0
#pragma unroll
    for (int kc = 0; kc < 3; ++kc) {
        float2 v = rp2[2 * kc + hi];
        float xv[2] = {v.x, v.y};
#pragma unroll
        for (int t = 0; t < 2; ++t) {
            float x = fminf(fmaxf(xv[t], 0.0f), 1.0f);
            float g = cospif(x);
            ng[2 * kc + t] = (g < 0.0f) ? 1.0f : 0.0f;
            lg[2 * kc + t] = fmaxf(logf(fabsf(g)), -1.0e30f);
        }
    }

    // ---- D = A x B + C chained over 3 K-chunks (K=12; j>=12 pad contributes 0) ----
    // B[k][n] = (k <= n) ? 1 : 0 ; lane holds n = m, k follows the same VGPR/half
    // pattern as A's K (VGPR0 -> k0/k2, VGPR1 -> k1/k3).
    v8f accL = {};  // sum of logs
    v8f accS = {};  // count of negative signs
#pragma unroll
    for (int kc = 0; kc < 3; ++kc) {
        const int k0 = 4 * kc + 2 * hi;
        v2f a; a.x = lg[2 * kc]; a.y = lg[2 * kc + 1];
        v2f c; c.x = ng[2 * kc]; c.y = ng[2 * kc + 1];
        v2f b;
        b.x = (k0 + 0 <= m) ? 1.0f : 0.0f;
        b.y = (k0 + 1 <= m) ? 1.0f : 0.0f;
        accL = __builtin_amdgcn_wmma_f32_16x16x4_f32(false, a, false, b,
                                                     (short)0, accL, false, false);
        accS = __builtin_amdgcn_wmma_f32_16x16x4_f32(false, c, false, b,
                                                     (short)0, accS, false, false);
    }

    // ---- store: C/D layout VGPR r, lane l -> (M = r + 8*hi, N = l&15) ----
    const int col = m;            // output qubit index = N
    if (col < NQ) {
#pragma unroll
        for (int r = 0; r < 8; ++r) {
            const int orow = row0 + r + 8 * hi;
            const int ci   = (int)(accS[r] + 0.5f);       // exact small integer
            const float sg = (ci & 1) ? -1.0f : 1.0f;
            out[orow * NQ + col] = sg * expf(accL[r]);
        }
    }
}

extern "C" void kernel_launch(void* const* d_in, const int* in_sizes, int n_in,
                              void* d_out, int out_size, void* d_ws, size_t ws_size,
                              hipStream_t stream) {
    (void)in_sizes; (void)n_in; (void)d_ws; (void)ws_size; (void)out_size;
    const float* in = (const float*)d_in[0];
    float* out      = (float*)d_out;
    // 16 rows/wave * 8 waves/block = 128 rows/block ; 16384/128 = 128 blocks
    qlayer_wmma_kernel<<<dim3(BATCH / 128), dim3(256), 0, stream>>>(in, out);
}